// Model_75058848465363
// MI455X (gfx1250) — compile-verified
//
#include <hip/hip_runtime.h>

typedef float v2f __attribute__((ext_vector_type(2)));
typedef float v8f __attribute__((ext_vector_type(8)));

#define N_NODES 50000
#define N_EDGES 320000
#define IN_CH 512
#define OUT_CH 256
#define MROWS 32
#define LDS_STRIDE 516   // 516 % 64 == 4 -> conflict-free, 16B aligned rows

// ---------------------------------------------------------------------------
// Kernel 1: h = x @ W^T using V_WMMA_F32_16X16X4_F32 (full fp32 precision)
// Block: 256 threads (8 waves). Each block: 32 nodes x 256 out-channels.
// Wave w owns out-cols [32w, 32w+32): 2 M-tiles x 2 N-tiles of 16x16.
// ---------------------------------------------------------------------------
__global__ __launch_bounds__(256) void gcn_gemm(const float* __restrict__ x,
                                                const float* __restrict__ W,
                                                float* __restrict__ h) {
    __shared__ float lds[MROWS * LDS_STRIDE];
    const int t  = threadIdx.x;
    const int mb = blockIdx.x * MROWS;

    // Stage 32x512 fp32 tile of x into LDS (zero-fill OOB rows).
#pragma unroll
    for (int i = 0; i < 16; ++i) {
        int flat = t + i * 256;       // in float4 units
        int row  = flat >> 7;         // 128 float4 per row of 512 floats
        int c4   = flat & 127;
        float4 v = make_float4(0.f, 0.f, 0.f, 0.f);
        int gr = mb + row;
        if (gr < N_NODES) v = *(const float4*)(x + (size_t)gr * IN_CH + (size_t)c4 * 4);
        *(float4*)(&lds[row * LDS_STRIDE + c4 * 4]) = v;
    }
    __syncthreads();

    const int wave = t >> 5;
    const int lane = t & 31;
    const int half = (lane < 16) ? 0 : 2;   // K sub-offset per ISA A/B layout
    const int m    = lane & 15;
    const int nb   = wave * 32;

    v8f acc00 = {}; v8f acc01 = {}; v8f acc10 = {}; v8f acc11 = {};

    const float* w0 = W + (size_t)(nb + m) * IN_CH;        // N-tile 0 column row of W
    const float* w1 = W + (size_t)(nb + 16 + m) * IN_CH;   // N-tile 1
    const float* l0 = &lds[(m)       * LDS_STRIDE];        // M-tile 0 row
    const float* l1 = &lds[(16 + m)  * LDS_STRIDE];        // M-tile 1 row

#pragma unroll 4
    for (int k0 = 0; k0 < IN_CH; k0 += 4) {
        v2f a0 = *(const v2f*)(l0 + k0 + half);
        v2f a1 = *(const v2f*)(l1 + k0 + half);
        v2f b0 = *(const v2f*)(w0 + k0 + half);
        v2f b1 = *(const v2f*)(w1 + k0 + half);
        acc00 = __builtin_amdgcn_wmma_f32_16x16x4_f32(false, a0, false, b0, (short)0, acc00, false, false);
        acc01 = __builtin_amdgcn_wmma_f32_16x16x4_f32(false, a0, false, b1, (short)0, acc01, false, false);
        acc10 = __builtin_amdgcn_wmma_f32_16x16x4_f32(false, a1, false, b0, (short)0, acc10, false, false);
        acc11 = __builtin_amdgcn_wmma_f32_16x16x4_f32(false, a1, false, b1, (short)0, acc11, false, false);
    }

    // Store D: lane holds col = tile_base + (lane&15); VGPR v -> row v (+8 for hi lanes).
    const int rbase = (lane < 16) ? 0 : 8;
    const int col0  = nb + m;
    const int col1  = nb + 16 + m;
#pragma unroll
    for (int v = 0; v < 8; ++v) {
        int g0 = mb + rbase + v;        // M-tile 0
        int g1 = mb + 16 + rbase + v;   // M-tile 1
        if (g0 < N_NODES) {
            h[(size_t)g0 * OUT_CH + col0] = acc00[v];
            h[(size_t)g0 * OUT_CH + col1] = acc01[v];
        }
        if (g1 < N_NODES) {
            h[(size_t)g1 * OUT_CH + col0] = acc10[v];
            h[(size_t)g1 * OUT_CH + col1] = acc11[v];
        }
    }
}

// ---------------------------------------------------------------------------
// Kernel 2: out[n][c] = b[c]
// ---------------------------------------------------------------------------
__global__ __launch_bounds__(256) void gcn_bias(float* __restrict__ out,
                                                const float* __restrict__ b) {
    size_t i = (size_t)blockIdx.x * 256 + threadIdx.x;  // 50000 blocks * 256 == out_size
    out[i] = b[threadIdx.x & 255];
}

// ---------------------------------------------------------------------------
// Kernel 3: out[dst] += ew * h[src]   (one wave per edge; 8 ch per lane)
// h and out both fit in the 192MB L2 -> atomics run at L2 speed.
// ---------------------------------------------------------------------------
__global__ __launch_bounds__(256) void gcn_scatter(const float* __restrict__ h,
                                                   const int* __restrict__ ei,
                                                   const float* __restrict__ ew,
                                                   float* __restrict__ out) {
    int e = blockIdx.x * 8 + (threadIdx.x >> 5);
    if (e >= N_EDGES) return;
    int lane = threadIdx.x & 31;
    int s = ei[e];
    int d = ei[N_EDGES + e];
    float w = ew[e];
    const float4* hs = (const float4*)(h + (size_t)s * OUT_CH);
    float* od = out + (size_t)d * OUT_CH;
    int c4 = lane * 2;                       // two float4 = 8 channels per lane
    float4 v0 = hs[c4];
    float4 v1 = hs[c4 + 1];
    int c = c4 * 4;
    atomicAdd(od + c + 0, w * v0.x);
    atomicAdd(od + c + 1, w * v0.y);
    atomicAdd(od + c + 2, w * v0.z);
    atomicAdd(od + c + 3, w * v0.w);
    atomicAdd(od + c + 4, w * v1.x);
    atomicAdd(od + c + 5, w * v1.y);
    atomicAdd(od + c + 6, w * v1.z);
    atomicAdd(od + c + 7, w * v1.w);
}

extern "C" void kernel_launch(void* const* d_in, const int* in_sizes, int n_in,
                              void* d_out, int out_size, void* d_ws, size_t ws_size,
                              hipStream_t stream) {
    const float* x  = (const float*)d_in[0];   // [50000, 512]
    const int*   ei = (const int*)d_in[1];     // [2, 320000]
    const float* ew = (const float*)d_in[2];   // [320000]
    const float* W  = (const float*)d_in[3];   // [256, 512]
    const float* b  = (const float*)d_in[4];   // [256]
    float* out = (float*)d_out;                // [50000, 256]
    float* h   = (float*)d_ws;                 // scratch: 50000*256*4 = 51.2 MB

    gcn_gemm<<<(N_NODES + MROWS - 1) / MROWS, 256, 0, stream>>>(x, W, h);
    gcn_bias<<<N_NODES, 256, 0, stream>>>(out, b);
    gcn_scatter<<<(N_EDGES + 7) / 8, 256, 0, stream>>>(h, ei, ew, out);
}